// Attention_layer_44813688766807
// MI455X (gfx1250) — compile-verified
//
#include <hip/hip_runtime.h>
#include <hip/hip_bf16.h>
#include <math.h>

// ---------------- problem constants ----------------
#define NN 50000      // nodes
#define NE 1600000    // edges
#define FIN 256       // input features
#define OUT 64        // per-head output
#define NH 8          // heads
#define HO (NH*OUT)   // 512

typedef __attribute__((ext_vector_type(2))) float v2f;
typedef __attribute__((ext_vector_type(8))) float v8f;

// ---------------- kernel 0: transpose W: Ws[H][O][F] -> Wt[H][F][O] ----------
__global__ void k_wt(const float* __restrict__ Ws, float* __restrict__ Wt) {
    int idx = blockIdx.x * 256 + threadIdx.x;          // H*F*O = 131072
    if (idx >= NH * FIN * OUT) return;
    int h = idx >> 14;                                 // F*O = 16384
    int rem = idx & 16383;
    int f = rem >> 6;
    int o = rem & 63;
    Wt[idx] = Ws[(h << 14) + (o << 8) + f];            // O*F = 16384, F = 256
}

// ---------------- kernel 1: t[h][n][o] = sum_f X[n][f] * W[h][o][f] ----------
// block = 256 threads = 8 waves; wave w == head w; block covers 16 node rows.
__global__ void k_gemm(const float* __restrict__ x, const float* __restrict__ Wt,
                       float* __restrict__ t) {
    __shared__ float xs[16 * FIN];                     // 16 KB X tile, shared by heads
    const int n0 = blockIdx.x * 16;
    const float* xsrc = x + (size_t)n0 * FIN;          // 16 contiguous rows
    for (int i = threadIdx.x; i < 16 * FIN; i += 256) xs[i] = xsrc[i];
    __syncthreads();

    const int h     = threadIdx.x >> 5;                // wave id = head
    const int lane  = threadIdx.x & 31;
    const int lhalf = lane >> 4;                       // 0: K0/K1 half, 1: K2/K3 half
    const int l16   = lane & 15;
    const float* wh = Wt + ((size_t)h << 14);          // [F][O]

    v8f c0 = {}, c1 = {}, c2 = {}, c3 = {};
    for (int k0 = 0; k0 < FIN; k0 += 4) {
        const int ka = k0 + 2 * lhalf;
        // A fragment 16x4 f32: lanes 0-15 rows, K = ka, ka+1 (ISA layout)
        v2f a;
        a.x = xs[l16 * FIN + ka];
        a.y = xs[l16 * FIN + ka + 1];
        // B fragments 4x16 f32, mirrored layout: lane half selects K pair
        const float* wk = wh + (size_t)ka * OUT;
        v2f b0, b1, b2, b3;
        b0.x = wk[l16];          b0.y = wk[OUT + l16];
        b1.x = wk[16 + l16];     b1.y = wk[OUT + 16 + l16];
        b2.x = wk[32 + l16];     b2.y = wk[OUT + 32 + l16];
        b3.x = wk[48 + l16];     b3.y = wk[OUT + 48 + l16];
        c0 = __builtin_amdgcn_wmma_f32_16x16x4_f32(false, a, false, b0, (short)0, c0, false, false);
        c1 = __builtin_amdgcn_wmma_f32_16x16x4_f32(false, a, false, b1, (short)0, c1, false, false);
        c2 = __builtin_amdgcn_wmma_f32_16x16x4_f32(false, a, false, b2, (short)0, c2, false, false);
        c3 = __builtin_amdgcn_wmma_f32_16x16x4_f32(false, a, false, b3, (short)0, c3, false, false);
    }
    // C/D layout: VGPR r -> row r + 8*lhalf, col = l16 (+16 per N tile)
    float* trow = t + ((size_t)h * NN + n0) * OUT;
    const int mb = 8 * lhalf;
#pragma unroll
    for (int r = 0; r < 8; ++r) {
        const int row = r + mb;
        trow[row * OUT + l16]      = c0[r];
        trow[row * OUT + 16 + l16] = c1[r];
        trow[row * OUT + 32 + l16] = c2[r];
        trow[row * OUT + 48 + l16] = c3[r];
    }
}

// ---------------- kernel 2: Al/Ar per (head, node) -------------------------
// block = 256 = 8 waves; block b covers node b, wave = head.
__global__ void k_alar(const float* __restrict__ t, const float* __restrict__ As,
                       const float* __restrict__ As_bias,
                       float* __restrict__ Al, float* __restrict__ Ar) {
    const int n = blockIdx.x;
    const int h = threadIdx.x >> 5;
    const int lane = threadIdx.x & 31;
    const float2 tv = *(const float2*)(t + ((size_t)h * NN + n) * OUT + 2 * lane);
    const float2 al = *(const float2*)(As + h * 2 * OUT + 2 * lane);
    const float2 ar = *(const float2*)(As + h * 2 * OUT + OUT + 2 * lane);
    float a = tv.x * al.x + tv.y * al.y;
    float r = tv.x * ar.x + tv.y * ar.y;
#pragma unroll
    for (int m = 16; m > 0; m >>= 1) {
        a += __shfl_xor(a, m, 32);
        r += __shfl_xor(r, m, 32);
    }
    if (lane == 0) {
        Al[(size_t)h * NN + n] = a + As_bias[h * 2 + 0];
        Ar[(size_t)h * NN + n] = r + As_bias[h * 2 + 1];
    }
}

// ---------------- CSR build -------------------------------------------------
__global__ void k_count(const int* __restrict__ erow, int* __restrict__ deg) {
    int e = blockIdx.x * 256 + threadIdx.x;
    if (e < NE) atomicAdd(&deg[erow[e]], 1);
}

// per-256-chunk inclusive scan; writes row_ptr[i+1] and block sums
__global__ void k_scan_block(const int* __restrict__ deg, int* __restrict__ row_ptr,
                             int* __restrict__ bsums) {
    __shared__ int s[256];
    const int tid = threadIdx.x;
    const int idx = blockIdx.x * 256 + tid;
    int v = (idx < NN) ? deg[idx] : 0;
    s[tid] = v;
    __syncthreads();
    for (int off = 1; off < 256; off <<= 1) {
        int u = (tid >= off) ? s[tid - off] : 0;
        __syncthreads();
        s[tid] += u;
        __syncthreads();
    }
    if (idx < NN) row_ptr[idx + 1] = s[tid];
    if (tid == 255) bsums[blockIdx.x] = s[255];
}

// single block: exclusive scan of block sums
__global__ void k_scan_sums(const int* __restrict__ bsums, int* __restrict__ boffs,
                            int nblk) {
    __shared__ int s[256];
    const int tid = threadIdx.x;
    int v = (tid < nblk) ? bsums[tid] : 0;
    s[tid] = v;
    __syncthreads();
    for (int off = 1; off < 256; off <<= 1) {
        int u = (tid >= off) ? s[tid - off] : 0;
        __syncthreads();
        s[tid] += u;
        __syncthreads();
    }
    boffs[tid] = s[tid] - v;   // exclusive
}

__global__ void k_scan_add(int* __restrict__ row_ptr, const int* __restrict__ boffs) {
    const int idx = blockIdx.x * 256 + threadIdx.x;
    if (idx < NN) row_ptr[idx + 1] += boffs[blockIdx.x];
    if (idx == 0) row_ptr[0] = 0;
}

__global__ void k_cursor(const int* __restrict__ row_ptr, int* __restrict__ cur) {
    const int i = blockIdx.x * 256 + threadIdx.x;
    if (i < NN) cur[i] = row_ptr[i];
}

__global__ void k_fill(const int* __restrict__ erow, const int* __restrict__ ecol,
                       int* __restrict__ cur, int* __restrict__ csr_col) {
    const int e = blockIdx.x * 256 + threadIdx.x;
    if (e >= NE) return;
    const int pos = atomicAdd(&cur[erow[e]], 1);
    csr_col[pos] = ecol[e];
}

// deterministic content: sort each row's neighbor list (insertion sort, avg deg 32)
__global__ void k_sort(const int* __restrict__ row_ptr, int* __restrict__ csr_col) {
    const int i = blockIdx.x * 256 + threadIdx.x;
    if (i >= NN) return;
    const int rs = row_ptr[i], re = row_ptr[i + 1];
    for (int a = rs + 1; a < re; ++a) {
        const int v = csr_col[a];
        int b = a - 1;
        while (b >= rs && csr_col[b] > v) { csr_col[b + 1] = csr_col[b]; --b; }
        csr_col[b + 1] = v;
    }
}

// ---------------- kernel 4: per-node softmax + aggregate + ELU -------------
// block = node i (256 thr = 8 waves); wave = head. No atomics; t gathers hit L2.
__global__ void k_agg(const int* __restrict__ row_ptr, const int* __restrict__ csr_col,
                      const float* __restrict__ Al, const float* __restrict__ Ar,
                      const float* __restrict__ t, const float* __restrict__ Ws_bias,
                      float* __restrict__ out) {
    const int i    = blockIdx.x;
    const int h    = threadIdx.x >> 5;
    const int lane = threadIdx.x & 31;
    const int rs = row_ptr[i];
    const int re = row_ptr[i + 1];
    const float ali = Al[(size_t)h * NN + i];
    const float* arh = Ar + (size_t)h * NN;
    const float* th  = t + (size_t)h * NN * OUT;

    // pass 1: row max
    float m = -__builtin_inff();
    for (int e = rs + lane; e < re; e += 32)
        m = fmaxf(m, ali + arh[csr_col[e]]);
#pragma unroll
    for (int d = 16; d > 0; d >>= 1) m = fmaxf(m, __shfl_xor(m, d, 32));

    // pass 2: sum of exp (prefetch t rows for pass 3)
    float s = 0.f;
    for (int e = rs + lane; e < re; e += 32) {
        const int j = csr_col[e];
        s += __expf(ali + arh[j] - m);
        __builtin_prefetch(th + (size_t)j * OUT, 0, 1);   // global_prefetch_b8
    }
#pragma unroll
    for (int d = 16; d > 0; d >>= 1) s += __shfl_xor(s, d, 32);

    // pass 3: acc = sum_j exp(score_j - m) * t[h][j][:], lanes cover 64 feats
    float2 acc = make_float2(0.f, 0.f);
    for (int base = rs; base < re; base += 32) {
        const int e = base + lane;
        float exl = 0.f;
        int jl = 0;
        if (e < re) {
            jl = csr_col[e];
            exl = __expf(ali + arh[jl] - m);
        }
        const int cnt = min(32, re - base);
        for (int k = 0; k < cnt; ++k) {
            const float w = __shfl(exl, k, 32);
            const int j   = __shfl(jl, k, 32);
            const float2 tv = *(const float2*)(th + (size_t)j * OUT + 2 * lane);
            acc.x += w * tv.x;
            acc.y += w * tv.y;
        }
    }
    const float inv = (re > rs) ? (1.0f / s) : 0.f;
    const float2 wb = *(const float2*)(Ws_bias + h * OUT + 2 * lane);
    float ox = acc.x * inv + wb.x;
    float oy = acc.y * inv + wb.y;
    ox = (ox > 0.f) ? ox : (__expf(ox) - 1.0f);   // ELU
    oy = (oy > 0.f) ? oy : (__expf(oy) - 1.0f);
    *(float2*)(out + (size_t)i * HO + h * OUT + 2 * lane) = make_float2(ox, oy);
}

// ---------------- launch ----------------------------------------------------
extern "C" void kernel_launch(void* const* d_in, const int* in_sizes, int n_in,
                              void* d_out, int out_size, void* d_ws, size_t ws_size,
                              hipStream_t stream) {
    (void)in_sizes; (void)n_in; (void)out_size; (void)ws_size;
    const float* x        = (const float*)d_in[0];
    const int*   erow     = (const int*)d_in[1];
    const int*   ecol     = (const int*)d_in[2];
    const float* Ws       = (const float*)d_in[3];
    const float* Ws_bias  = (const float*)d_in[4];
    const float* As       = (const float*)d_in[5];
    const float* As_bias  = (const float*)d_in[6];
    float* out = (float*)d_out;

    // workspace carve-up (256 B aligned)
    char* ws = (char*)d_ws;
    size_t off = 0;
    auto carve = [&](size_t bytes) {
        void* p = ws + off;
        off += (bytes + 255) & ~(size_t)255;
        return p;
    };
    float* t       = (float*)carve((size_t)NH * NN * OUT * 4); // 102.4 MB (L2-resident)
    float* Wt      = (float*)carve((size_t)NH * FIN * OUT * 4);
    float* Al      = (float*)carve((size_t)NH * NN * 4);
    float* Ar      = (float*)carve((size_t)NH * NN * 4);
    int*   deg     = (int*)carve((size_t)NN * 4);
    int*   row_ptr = (int*)carve((size_t)(NN + 1) * 4);
    int*   bsums   = (int*)carve(256 * 4);
    int*   boffs   = (int*)carve(256 * 4);
    int*   cur     = (int*)carve((size_t)NN * 4);
    int*   csr_col = (int*)carve((size_t)NE * 4);

    const int nodeBlk  = (NN + 255) / 256;   // 196
    const int edgeBlk  = (NE + 255) / 256;   // 6250

    // dense transform (WMMA)
    k_wt<<<(NH * FIN * OUT + 255) / 256, 256, 0, stream>>>(Ws, Wt);
    k_gemm<<<NN / 16, 256, 0, stream>>>(x, Wt, t);
    k_alar<<<NN, 256, 0, stream>>>(t, As, As_bias, Al, Ar);

    // CSR build
    hipMemsetAsync(deg, 0, (size_t)NN * 4, stream);
    k_count<<<edgeBlk, 256, 0, stream>>>(erow, deg);
    k_scan_block<<<nodeBlk, 256, 0, stream>>>(deg, row_ptr, bsums);
    k_scan_sums<<<1, 256, 0, stream>>>(bsums, boffs, nodeBlk);
    k_scan_add<<<nodeBlk, 256, 0, stream>>>(row_ptr, boffs);
    k_cursor<<<nodeBlk, 256, 0, stream>>>(row_ptr, cur);
    k_fill<<<edgeBlk, 256, 0, stream>>>(erow, ecol, cur, csr_col);
    k_sort<<<nodeBlk, 256, 0, stream>>>(row_ptr, csr_col);

    // edge softmax + aggregation + ELU
    k_agg<<<NN, 256, 0, stream>>>(row_ptr, csr_col, Al, Ar, t, Ws_bias, out);
}